// TSELKBlock_no_tail_norm_64613488001621
// MI455X (gfx1250) — compile-verified
//
#include <hip/hip_runtime.h>
#include <hip/hip_bf16.h>

typedef float v2f __attribute__((ext_vector_type(2)));
typedef float v8f __attribute__((ext_vector_type(8)));

#define TN      300000
#define INC     128
#define NB      4
#define GXC     50
#define GYC     50
#define GZC     4
#define STRIDEC 8
#define NCELLS  (NB*GXC*GYC*GZC)      // 40000
#define GCOLS   (2*INC+1)             // 257
#define GRID_ELEMS ((size_t)NCELLS*GCOLS)
#define AGG_ELEMS  ((size_t)NCELLS*(2*INC))

// ---------------- kernel 0: zero the scatter grid ----------------
__global__ void k_zero(float* __restrict__ p, int n) {
    for (int i = blockIdx.x * blockDim.x + threadIdx.x; i < n; i += gridDim.x * blockDim.x)
        p[i] = 0.0f;
}

// ---------------- kernel 1: WMMA GEMM + LayerNorm + sincos + scatter ----------------
// One wave32 computes h for 16 points x 128 channels via V_WMMA_F32_16X16X4_F32.
__global__ __launch_bounds__(256) void k_gemm_ln_scatter(
    const float* __restrict__ x, const int* __restrict__ coords,
    const float* __restrict__ Wm, const float* __restrict__ lng,
    const float* __restrict__ lnb, const float* __restrict__ Wp,
    float* __restrict__ gridBuf)
{
    __shared__ float sW[INC * INC];                 // 64 KB: W_mix row-major
    const int tid = threadIdx.x;

    // cooperative load of W_mix into LDS (float4)
    {
        const float4* src = (const float4*)Wm;
        float4* dst = (float4*)sW;
        #pragma unroll
        for (int i = 0; i < (INC * INC / 4) / 256; ++i)
            dst[tid + i * 256] = src[tid + i * 256];
    }
    __syncthreads();

    const int wave = blockIdx.x * 8 + (tid >> 5);
    if (wave * 16 >= TN) return;                    // uniform per-wave tail skip

    const int lane = tid & 31;
    const int half = lane >> 4;                     // 0: K lo / point M<8 ; 1: K hi / point M>=8
    const int ln16 = lane & 15;
    const int base = wave * 16;

    // ---- preload A fragments: 16x4 f32 tiles, lane ln16 row, halves split K ----
    v2f a[32];
    {
        const float* xr = x + (size_t)(base + ln16) * INC + 2 * half;
        #pragma unroll
        for (int ks = 0; ks < 32; ++ks)
            a[ks] = *(const v2f*)(xr + ks * 4);
    }

    // ---- per-point coords / voxel id (lanes 0..15 own point base+ln16) ----
    float cxf, cyf, czf; int vidL;
    {
        const int* cr = coords + (size_t)(base + ln16) * 4;
        int ci = cr[0], cj = cr[1], ck = cr[2], cb = cr[3];
        cxf = (float)ci; cyf = (float)cj; czf = (float)ck;
        vidL = ((cb * GXC + ci / STRIDEC) * GYC + cj / STRIDEC) * GZC + ck / STRIDEC;
    }
    if (lane < 16)                                   // count column: one atomic per point
        unsafeAtomicAdd(&gridBuf[(size_t)vidL * GCOLS + 2 * INC], 1.0f);

    // ---- GEMM: h = x @ W_mix^T, 8 j-tiles x 32 k-steps of f32 WMMA ----
    v8f acc[8];
    #pragma unroll
    for (int jt = 0; jt < 8; ++jt) {
        v8f cacc = {};
        const float* wb = &sW[(jt * 16 + ln16) * INC + 2 * half];   // B[k,n] = W_mix[n,k]
        #pragma unroll
        for (int ks = 0; ks < 32; ++ks) {
            v2f b = *(const v2f*)(wb + ks * 4);
            cacc = __builtin_amdgcn_wmma_f32_16x16x4_f32(
                false, a[ks], false, b, (short)0, cacc, false, false);
        }
        acc[jt] = cacc;
    }

    // ---- LayerNorm stats: reduce 128 channels = 8 jt-tiles x 16 lanes (per half) ----
    float mu[8], ri[8];
    #pragma unroll
    for (int r = 0; r < 8; ++r) {
        float s = 0.f, q = 0.f;
        #pragma unroll
        for (int jt = 0; jt < 8; ++jt) { float v = acc[jt][r]; s += v; q += v * v; }
        #pragma unroll
        for (int m = 1; m < 16; m <<= 1) {          // butterfly stays inside 16-lane half
            s += __shfl_xor(s, m, 32);
            q += __shfl_xor(q, m, 32);
        }
        float mean = s * (1.0f / INC);
        float var  = q * (1.0f / INC) - mean * mean;
        mu[r] = mean;
        ri[r] = rsqrtf(var + 1e-6f);
    }

    // ---- normalize, modulate by sin/cos(pw), scatter-add into grid ----
    #pragma unroll
    for (int jt = 0; jt < 8; ++jt) {
        const int c0 = jt * 16 + ln16;               // channel this lane holds
        const float gg = lng[c0], bb = lnb[c0];
        const float w0 = Wp[c0], w1 = Wp[INC + c0], w2 = Wp[2 * INC + c0];
        #pragma unroll
        for (int r = 0; r < 8; ++r) {
            const int src = r + 8 * half;            // lane holding point (r + 8*half)'s coords
            float pcx = __shfl(cxf, src, 32);
            float pcy = __shfl(cyf, src, 32);
            float pcz = __shfl(czf, src, 32);
            int   pv  = __shfl(vidL, src, 32);
            float hn  = (acc[jt][r] - mu[r]) * ri[r] * gg + bb;
            float pw  = pcx * w0 + pcy * w1 + pcz * w2;
            float sv, cv;
            __sincosf(pw, &sv, &cv);
            float* cell = gridBuf + (size_t)pv * GCOLS;
            unsafeAtomicAdd(cell + c0,       hn * cv);
            unsafeAtomicAdd(cell + INC + c0, hn * sv);
        }
    }
}

// ---------------- kernel 2: 3x3x3 neighbor sum + normalize (grid is L2-resident) ----------------
__global__ __launch_bounds__(256) void k_neighbor(
    const float* __restrict__ gridBuf, float* __restrict__ agg)
{
    __shared__ int   nOff[27];
    __shared__ float cntS;
    const int cell = blockIdx.x;
    const int tid  = threadIdx.x;                    // channel 0..255
    int t  = cell;
    const int gz = t % GZC; t /= GZC;
    const int gy = t % GYC; t /= GYC;
    const int gx = t % GXC;
    const int bb = t / GXC;

    if (tid == 0) cntS = 0.0f;
    __syncthreads();
    if (tid < 27) {
        int dx = tid / 9 - 1, dy = (tid / 3) % 3 - 1, dz = tid % 3 - 1;
        int nx = gx + dx, ny = gy + dy, nz = gz + dz;
        int off = -1;
        if (nx >= 0 && nx < GXC && ny >= 0 && ny < GYC && nz >= 0 && nz < GZC)
            off = ((bb * GXC + nx) * GYC + ny) * GZC + nz;
        nOff[tid] = off;
        if (off >= 0) atomicAdd(&cntS, gridBuf[(size_t)off * GCOLS + 2 * INC]);
    }
    __syncthreads();

    float s = 0.0f;
    #pragma unroll
    for (int k = 0; k < 27; ++k) {
        int o = nOff[k];
        if (o >= 0) s += gridBuf[(size_t)o * GCOLS + tid];
    }
    agg[(size_t)cell * (2 * INC) + tid] = s / fmaxf(cntS, 1e-12f);
}

// ---------------- kernel 3: gather agg[vid], combine with cos/sin ----------------
__global__ __launch_bounds__(128) void k_gather(
    const int* __restrict__ coords, const float* __restrict__ Wp,
    const float* __restrict__ agg, float* __restrict__ out)
{
    const int n = blockIdx.x;
    const int c = threadIdx.x;
    const int* cr = coords + (size_t)n * 4;
    const int ci = cr[0], cj = cr[1], ck = cr[2], cb = cr[3];
    const int vid = ((cb * GXC + ci / STRIDEC) * GYC + cj / STRIDEC) * GZC + ck / STRIDEC;
    float pw = (float)ci * Wp[c] + (float)cj * Wp[INC + c] + (float)ck * Wp[2 * INC + c];
    float sv, cv;
    __sincosf(pw, &sv, &cv);
    const float* row = agg + (size_t)vid * (2 * INC);
    out[(size_t)n * INC + c] = row[c] * cv + row[INC + c] * sv;
}

extern "C" void kernel_launch(void* const* d_in, const int* in_sizes, int n_in,
                              void* d_out, int out_size, void* d_ws, size_t ws_size,
                              hipStream_t stream) {
    const float* x      = (const float*)d_in[0];
    const int*   coords = (const int*)  d_in[1];
    const float* Wm     = (const float*)d_in[2];
    const float* lng    = (const float*)d_in[3];
    const float* lnb    = (const float*)d_in[4];
    const float* Wp     = (const float*)d_in[5];
    (void)in_sizes; (void)n_in; (void)out_size; (void)ws_size;

    float* gridBuf = (float*)d_ws;
    float* agg     = gridBuf + GRID_ELEMS;
    float* out     = (float*)d_out;

    k_zero<<<2048, 256, 0, stream>>>(gridBuf, (int)GRID_ELEMS);

    const int nWaves  = TN / 16;                     // 18750
    const int nBlocks = (nWaves + 7) / 8;            // 8 waves / 256-thread block
    k_gemm_ln_scatter<<<nBlocks, 256, 0, stream>>>(x, coords, Wm, lng, lnb, Wp, gridBuf);

    k_neighbor<<<NCELLS, 256, 0, stream>>>(gridBuf, agg);

    k_gather<<<TN, 128, 0, stream>>>(coords, Wp, agg, out);
}